// _H3Preprocessing_47278999994556
// MI455X (gfx1250) — compile-verified
//
#include <hip/hip_runtime.h>
#include <hip/hip_bf16.h>
#include <stdint.h>

// H3 index decode: [N] int64 -> [N,19] float (mode, edge, res, base, 15 digits).
// Memory-bound streaming kernel: ~8B in + 76B out per row (~704 MB @ 23.3 TB/s
// => ~30us floor). No matrix math -> no WMMA; the CDNA5 path that matters is
// LDS staging + async LDS->global b128 stores with non-temporal cache hints.

#define RPB     256                 // rows per block (== blockDim.x)
#define ROW_F   19                  // output floats per row
#define TILE_F  (RPB * ROW_F)       // 4864 floats = 19456 bytes of LDS
#define TILE_V4 (TILE_F / 4)        // 1216 16-byte chunks (19456 % 16 == 0)
#define V4_ITERS ((TILE_V4 + RPB - 1) / RPB)   // 5

#ifndef USE_ASYNC_STORE
#define USE_ASYNC_STORE 1           // gfx1250 global_store_async_from_lds_b128
#endif

typedef float v4f __attribute__((ext_vector_type(4)));

__global__ __launch_bounds__(RPB) void h3_decode_kernel(
    const unsigned long long* __restrict__ v,
    float* __restrict__ out,
    int n)
{
    __shared__ __align__(16) float tile[TILE_F];

    const int tid  = threadIdx.x;
    const int row0 = blockIdx.x * RPB;
    const int row  = row0 + tid;

    // ---- load one int64 per lane (coalesced b64, read-once -> NT) ----------
    unsigned long long x = 0ull;
    if (row < n) x = __builtin_nontemporal_load(v + row);

    // ---- decode bit-fields -------------------------------------------------
    const unsigned hi   = (unsigned)(x >> 32);
    const unsigned mode = (hi >> 27) & 0xFu;   // bits 62:59
    const unsigned edge = (hi >> 24) & 0x7u;   // bits 58:56
    const unsigned res  = (hi >> 20) & 0xFu;   // bits 55:52
    const unsigned base = (hi >> 13) & 0x7Fu;  // bits 51:45

    // Row-major LDS staging. Per-lane base stride = 19 floats; 19 is coprime
    // with 64 banks, so the ds_store_b32 pattern below is conflict-free.
    float* r = &tile[tid * ROW_F];
    r[0] = (float)mode;
    r[1] = (float)edge;
    r[2] = (float)res;
    r[3] = (float)base;
#pragma unroll
    for (int k = 0; k < 15; ++k) {
        unsigned c = (unsigned)(x >> (3 * (14 - k))) & 0x7u;
        r[4 + k] = (float)(((unsigned)k < res) ? c : 7u);   // pad beyond res
    }
    __syncthreads();   // s_wait_dscnt 0 + barrier: LDS tile complete

    // ---- write out: contiguous 19456B per block, 16B aligned ---------------
    float* gout = out + (size_t)row0 * ROW_F;
    const int rows = min(RPB, n - row0);

    if (rows == RPB) {
#if USE_ASYNC_STORE
        // Async DMA the tile straight from LDS to global memory (bypasses the
        // ds_load_b128 + VGPR round-trip). 1216 b128 chunks, NT store hint.
#pragma unroll
        for (int i = 0; i < V4_ITERS; ++i) {
            int idx = tid + i * RPB;
            if (idx < TILE_V4) {
                unsigned lds_addr =
                    (unsigned)(uintptr_t)(__attribute__((address_space(3))) float*)&tile[idx * 4];
                unsigned goff = (unsigned)(idx * 16);   // byte offset from gout
                asm volatile(
                    "global_store_async_from_lds_b128 %0, %1, %2 th:TH_STORE_NT"
                    :
                    : "v"(goff), "v"(lds_addr), "s"(gout)
                    : "memory");
            }
        }
        // Drain ASYNCcnt before the workgroup releases its LDS.
#if __has_builtin(__builtin_amdgcn_s_wait_asynccnt)
        __builtin_amdgcn_s_wait_asynccnt(0);
#else
        asm volatile("s_wait_asynccnt 0x0" ::: "memory");
#endif
#else
        // Fallback: ds_load_b128 + non-temporal global_store_b128.
#pragma unroll
        for (int i = 0; i < V4_ITERS; ++i) {
            int idx = tid + i * RPB;
            if (idx < TILE_V4) {
                v4f val = *(const v4f*)&tile[idx * 4];
                __builtin_nontemporal_store(val, (v4f*)gout + idx);
            }
        }
#endif
    } else {
        // Generic tail block (not taken for N = 2^23): coalesced b32 NT stores.
        const int nflt = rows * ROW_F;
        for (int i = tid; i < nflt; i += RPB) {
            __builtin_nontemporal_store(tile[i], gout + i);
        }
    }
}

extern "C" void kernel_launch(void* const* d_in, const int* in_sizes, int n_in,
                              void* d_out, int out_size, void* d_ws, size_t ws_size,
                              hipStream_t stream) {
    (void)n_in; (void)out_size; (void)d_ws; (void)ws_size;
    // Reference input dtype is jnp.int64 -> 8-byte integers.
    const unsigned long long* v = (const unsigned long long*)d_in[0];
    float* out = (float*)d_out;                  // non-bf16 output -> float
    const int n = in_sizes[0];                   // 8,388,608 rows
    const int blocks = (n + RPB - 1) / RPB;      // 32768 blocks, divides evenly
    h3_decode_kernel<<<blocks, RPB, 0, stream>>>(v, out, n);
}